// AdaptiveSNN_19954418057508
// MI455X (gfx1250) — compile-verified
//
#include <hip/hip_runtime.h>

// CDNA5 / gfx1250 fused SNN kernel.
// Inputs: x[16384,784] f32, W1[128,784], b1[128], W2[10,128], b2[10], beta1, beta2
// Output: spk2_rec[25,16384,10] f32 ++ mem2_rec[25,16384,10] f32

typedef __attribute__((ext_vector_type(2))) float v2f;
typedef __attribute__((ext_vector_type(4))) float v4f;
typedef __attribute__((ext_vector_type(8))) float v8f;
typedef __attribute__((ext_vector_type(4))) int   v4i;

#define AS_GLOBAL __attribute__((address_space(1)))
#define AS_LDS    __attribute__((address_space(3)))

#if defined(__gfx1250__) && __has_builtin(__builtin_amdgcn_global_load_async_to_lds_b128) && __has_builtin(__builtin_amdgcn_s_wait_asynccnt)
#define HAVE_ASYNC_LDS 1
#else
#define HAVE_ASYNC_LDS 0
#endif

#define NUM_STEPS 25
#define B_TOTAL   16384
#define HID       128
#define NOUT      10
#define KDIM      784
#define CHUNK     28                 // 784 = 28 * 28
#define NCHUNK    (KDIM / CHUNK)     // 28
#define SEGS      (CHUNK / 4)        // 7 x b128 per row
#define W1TILE    (HID * CHUNK)      // 3584 floats = 14 KB
#define WAVES_PER_BLOCK 8
#define THREADS   (32 * WAVES_PER_BLOCK)

// Stage one 128 x 28 chunk of W1 into LDS (row-major, stride CHUNK floats).
// Threads 0..127 each copy one row (7 x 16B). Async on gfx1250, sync fallback otherwise.
__device__ __forceinline__ void stage_w1_chunk(const float* __restrict__ W1,
                                               float* dst, int chunk, int tid) {
    if (tid < HID) {
        const float* src = W1 + (size_t)tid * KDIM + chunk * CHUNK;
        float* d = dst + tid * CHUNK;
#if HAVE_ASYNC_LDS
        #pragma unroll
        for (int seg = 0; seg < SEGS; ++seg) {
            __builtin_amdgcn_global_load_async_to_lds_b128(
                (AS_GLOBAL v4i*)(src + seg * 4),
                (AS_LDS v4i*)(d + seg * 4),
                0, 0);
        }
#else
        #pragma unroll
        for (int seg = 0; seg < SEGS; ++seg) {
            *(v4f*)(d + seg * 4) = *(const v4f*)(src + seg * 4);
        }
#endif
    }
}

__global__ __launch_bounds__(THREADS) void snn_fused_kernel(
    const float* __restrict__ x,
    const float* __restrict__ W1,
    const float* __restrict__ b1,
    const float* __restrict__ W2,
    const float* __restrict__ b2,
    const float* __restrict__ pbeta1,
    const float* __restrict__ pbeta2,
    float* __restrict__ spk_out,
    float* __restrict__ mem_out)
{
    // 64 KB arena:
    //  phase 1: [0, 2*W1TILE) = double-buffered W1 chunk staging
    //  phase 2: 8 waves x (16x128) spike tiles in [k][m] layout
    __shared__ float smem[16384];

    const int tid  = threadIdx.x;
    const int lane = tid & 31;
    const int wave = tid >> 5;
    const int half = lane >> 4;   // 0: lanes 0-15, 1: lanes 16-31
    const int lm   = lane & 15;
    const int row0 = (blockIdx.x * WAVES_PER_BLOCK + wave) * 16;

    const float beta1 = fminf(fmaxf(pbeta1[0], 0.0f), 1.0f);
    const float beta2 = fminf(fmaxf(pbeta2[0], 0.0f), 1.0f);

    // ---------------- Phase 1: cur1 = x @ W1^T + b1 (16 rows x 128 cols per wave)
    // C/D layout: element j of cur1[t] = (row = j + 8*half, col = 16*t + lm)
    v8f cur1[8];
    #pragma unroll
    for (int t = 0; t < 8; ++t) {
        float bv = b1[t * 16 + lm];
        cur1[t] = (v8f){bv, bv, bv, bv, bv, bv, bv, bv};
    }

    // A (16x4 f32): half 0 holds K = k0,k0+1 ; half 1 holds K = k0+2,k0+3 ; M = lm
    const float* xrow = x + (size_t)(row0 + lm) * KDIM + 2 * half;

    float* buf0 = smem;
    float* buf1 = smem + W1TILE;

    stage_w1_chunk(W1, buf0, 0, tid);

    for (int c = 0; c < NCHUNK; ++c) {
        float* curbuf = (c & 1) ? buf1 : buf0;
        float* nxtbuf = (c & 1) ? buf0 : buf1;
        if (c + 1 < NCHUNK) {
            stage_w1_chunk(W1, nxtbuf, c + 1, tid);
#if HAVE_ASYNC_LDS
            __builtin_amdgcn_s_wait_asynccnt(SEGS);  // chunk c complete, c+1 in flight
#endif
        } else {
#if HAVE_ASYNC_LDS
            __builtin_amdgcn_s_wait_asynccnt(0);
#endif
        }
        __syncthreads();   // chunk c visible to all 8 waves

        #pragma unroll
        for (int ks = 0; ks < SEGS; ++ks) {
            v2f a = *(const v2f*)(xrow + c * CHUNK + ks * 4);
            #pragma unroll
            for (int t = 0; t < 8; ++t) {
                // B (4x16): B[k][n] = W1[n][k]; lane n = 16*t+lm, K pair per half.
                v2f bw = *(const v2f*)(curbuf + (t * 16 + lm) * CHUNK + ks * 4 + 2 * half);
                cur1[t] = __builtin_amdgcn_wmma_f32_16x16x4_f32(
                    false, a, false, bw, (short)0, cur1[t], false, false);
            }
        }
        __syncthreads();   // all readers done before buffer is overwritten
    }

    // ---------------- Preload W2 B-fragments (N padded 10 -> 16), constant over steps.
    v2f w2f[32];
    #pragma unroll
    for (int kk = 0; kk < 32; ++kk) {
        if (lm < NOUT) {
            w2f[kk] = *(const v2f*)(W2 + lm * HID + 4 * kk + 2 * half);
        } else {
            w2f[kk] = (v2f){0.0f, 0.0f};
        }
    }

    // ---------------- Phase 2: 25-step LIF recurrence; spk1 transposed via LDS.
    float* myLds = smem + wave * (16 * HID);   // per-wave 16x128 tile, [k][m] layout

    v8f mem1[8];
    #pragma unroll
    for (int t = 0; t < 8; ++t) mem1[t] = (v8f){0, 0, 0, 0, 0, 0, 0, 0};
    v8f mem2 = (v8f){0, 0, 0, 0, 0, 0, 0, 0};
    const float bias2 = (lm < NOUT) ? b2[lm] : 0.0f;

    for (int step = 0; step < NUM_STEPS; ++step) {
        // Layer-1 membrane update + spike; write spikes to LDS in [k][m] layout.
        #pragma unroll
        for (int t = 0; t < 8; ++t) {
            float s[8];
            #pragma unroll
            for (int j = 0; j < 8; ++j) {
                float m = mem1[t][j];
                float reset = (m > 1.0f) ? 1.0f : 0.0f;        // spike of previous step
                float v = fmaf(beta1, m, cur1[t][j]) - reset;   // subtract-reset
                mem1[t][j] = v;
                s[j] = (v > 1.0f) ? 1.0f : 0.0f;                // Heaviside(v - thresh)
            }
            float* p = myLds + (t * 16 + lm) * 16 + 8 * half;   // rows contiguous
            *(v4f*)(p)     = (v4f){s[0], s[1], s[2], s[3]};
            *(v4f*)(p + 4) = (v4f){s[4], s[5], s[6], s[7]};
        }
        __syncthreads();

        // Layer-2 GEMM: cur2 = spk1 @ W2^T + b2 (K = 128, 32 WMMA k-steps)
        v8f c2 = (v8f){bias2, bias2, bias2, bias2, bias2, bias2, bias2, bias2};
        #pragma unroll
        for (int kk = 0; kk < 32; ++kk) {
            int k = 4 * kk + 2 * half;
            v2f a2;
            a2.x = myLds[k * 16 + lm];
            a2.y = myLds[(k + 1) * 16 + lm];
            c2 = __builtin_amdgcn_wmma_f32_16x16x4_f32(
                false, a2, false, w2f[kk], (short)0, c2, false, false);
        }
        __syncthreads();

        // Layer-2 membrane update + record spk2 / mem2.
        const size_t tbase = (size_t)step * B_TOTAL * NOUT;
        #pragma unroll
        for (int j = 0; j < 8; ++j) {
            float m = mem2[j];
            float reset = (m > 1.0f) ? 1.0f : 0.0f;
            float v = fmaf(beta2, m, c2[j]) - reset;
            mem2[j] = v;
            float sp = (v > 1.0f) ? 1.0f : 0.0f;
            if (lm < NOUT) {
                size_t o = tbase + (size_t)(row0 + j + 8 * half) * NOUT + lm;
                spk_out[o] = sp;
                mem_out[o] = v;
            }
        }
    }
}

extern "C" void kernel_launch(void* const* d_in, const int* in_sizes, int n_in,
                              void* d_out, int out_size, void* d_ws, size_t ws_size,
                              hipStream_t stream) {
    const float* x     = (const float*)d_in[0];
    const float* W1    = (const float*)d_in[1];
    const float* b1    = (const float*)d_in[2];
    const float* W2    = (const float*)d_in[3];
    const float* b2    = (const float*)d_in[4];
    const float* beta1 = (const float*)d_in[5];
    const float* beta2 = (const float*)d_in[6];

    float* spk = (float*)d_out;
    float* mem = spk + (size_t)NUM_STEPS * B_TOTAL * NOUT;

    dim3 grid(B_TOTAL / (16 * WAVES_PER_BLOCK));  // 128 blocks
    dim3 block(THREADS);                           // 256 threads = 8 waves
    hipLaunchKernelGGL(snn_fused_kernel, grid, block, 0, stream,
                       x, W1, b1, W2, b2, beta1, beta2, spk, mem);
}